// DynamicGenerator_17583596110050
// MI455X (gfx1250) — compile-verified
//
#include <hip/hip_runtime.h>

#define DT_STEP 0.1f
#define D_MODEL 256

typedef __attribute__((ext_vector_type(2))) float v2f;
typedef __attribute__((ext_vector_type(4))) float v4f;
typedef __attribute__((ext_vector_type(8))) float v8f;

// One wave32 handles a tile of 16 agents:
//   stage 1: st = tgt[row]@W_f2s + b ; u = x[row]@W_proj + b  (coalesced row reads,
//            per-lane register-resident weight slice, butterfly reduce)
//   dynamics: double-integrator (IS_UNI=0) or unicycle (IS_UNI=1)
//   stage 2: out[row] = next_state(16x4) @ W_s2f(4x256) + b via 16x V_WMMA_F32_16X16X4_F32
template <int IS_UNI>
__global__ __launch_bounds__(256) void agent_fused_kernel(
    const float* __restrict__ x, const float* __restrict__ tgt,
    const int* __restrict__ idx,
    const float* __restrict__ W_proj, const float* __restrict__ b_proj,
    const float* __restrict__ W_f2s,  const float* __restrict__ b_f2s,
    const float* __restrict__ W_s2f,  const float* __restrict__ b_s2f,
    float* __restrict__ out)
{
    const int lane = threadIdx.x & 31;
    const int hi   = lane >> 4;        // 0: low half (K=0,1 / M=r), 1: high half (K=2,3 / M=r+8)
    const int m    = lane & 15;
    const int wid  = (blockIdx.x << 3) + (threadIdx.x >> 5);   // global wave id
    const int j0   = wid << 4;         // first position in idx[] of this wave's 16-agent tile

    // ---- per-lane stage-1 weight slice: columns [lane*8, lane*8+8), kept in VGPRs ----
    const v4f* __restrict__ Wf4 = (const v4f*)W_f2s;   // W_f2s[c][0..3]
    const v2f* __restrict__ Wp2 = (const v2f*)W_proj;  // W_proj[c][0..1]
    v4f wf[8];
    v2f wp[8];
#pragma unroll
    for (int j = 0; j < 8; ++j) {
        wf[j] = Wf4[lane * 8 + j];
        wp[j] = Wp2[lane * 8 + j];
    }
    const float bf0 = b_f2s[0], bf1 = b_f2s[1], bf2 = b_f2s[2], bf3 = b_f2s[3];
    const float bp0 = b_proj[0], bp1 = b_proj[1];

    const int myrow = idx[j0 + m];     // lanes l and l+16 hold the same agent's row

    // ---- stage 1: build WMMA A-fragment (16x4 f32 layout) ----
    v2f A = (v2f)0.0f;
    for (int a = 0; a < 16; ++a) {
        const int row = __shfl(myrow, a, 32);
        const v4f* __restrict__ tr = (const v4f*)(tgt + (size_t)row * D_MODEL + lane * 8);
        const v4f* __restrict__ xr = (const v4f*)(x   + (size_t)row * D_MODEL + lane * 8);
        const v4f t0 = __builtin_nontemporal_load(tr);
        const v4f t1 = __builtin_nontemporal_load(tr + 1);
        const v4f x0 = __builtin_nontemporal_load(xr);
        const v4f x1 = __builtin_nontemporal_load(xr + 1);

        float s0 = 0.f, s1 = 0.f, s2 = 0.f, s3 = 0.f, p0 = 0.f, p1 = 0.f;
#pragma unroll
        for (int j = 0; j < 4; ++j) {
            const float tv = t0[j], xv = x0[j];
            s0 = fmaf(tv, wf[j].x, s0); s1 = fmaf(tv, wf[j].y, s1);
            s2 = fmaf(tv, wf[j].z, s2); s3 = fmaf(tv, wf[j].w, s3);
            p0 = fmaf(xv, wp[j].x, p0); p1 = fmaf(xv, wp[j].y, p1);
        }
#pragma unroll
        for (int j = 0; j < 4; ++j) {
            const float tv = t1[j], xv = x1[j];
            s0 = fmaf(tv, wf[4 + j].x, s0); s1 = fmaf(tv, wf[4 + j].y, s1);
            s2 = fmaf(tv, wf[4 + j].z, s2); s3 = fmaf(tv, wf[4 + j].w, s3);
            p0 = fmaf(xv, wp[4 + j].x, p0); p1 = fmaf(xv, wp[4 + j].y, p1);
        }

        // wave32 butterfly reduction: every lane ends with the full 6-vector
#pragma unroll
        for (int off = 16; off > 0; off >>= 1) {
            s0 += __shfl_xor(s0, off, 32);
            s1 += __shfl_xor(s1, off, 32);
            s2 += __shfl_xor(s2, off, 32);
            s3 += __shfl_xor(s3, off, 32);
            p0 += __shfl_xor(p0, off, 32);
            p1 += __shfl_xor(p1, off, 32);
        }
        s0 += bf0; s1 += bf1; s2 += bf2; s3 += bf3;
        p0 += bp0; p1 += bp1;

        float n0, n1, n2, n3;
        if (IS_UNI) {
            const float c = cosf(s2), sn = sinf(s2);
            n0 = fmaf(DT_STEP * s3, c,  s0);
            n1 = fmaf(DT_STEP * s3, sn, s1);
            n2 = fmaf(DT_STEP, p1, s2);
            n3 = fmaf(DT_STEP, p0, s3);
        } else {
            n0 = fmaf(DT_STEP, s2, s0);
            n1 = fmaf(DT_STEP, s3, s1);
            n2 = fmaf(DT_STEP, p0, s2);
            n3 = fmaf(DT_STEP, p1, s3);
        }
        // A-fragment (16x4 f32): lane<16 holds {K=0,K=1}, lane>=16 holds {K=2,K=3} of agent m
        if (m == a) {
            A.x = hi ? n2 : n0;
            A.y = hi ? n3 : n1;
        }
    }

    // ---- stage 2: D(16x16) = A(16x4) x B(4x16 slice of W_s2f), 16 tiles ----
    const float* __restrict__ Wr0 = W_s2f + (size_t)(2 * hi)     * D_MODEL;  // K = 2*hi
    const float* __restrict__ Wr1 = W_s2f + (size_t)(2 * hi + 1) * D_MODEL;  // K = 2*hi+1
#pragma unroll
    for (int t = 0; t < 16; ++t) {
        const int ncol = t * 16 + m;
        v2f B;
        B.x = Wr0[ncol];
        B.y = Wr1[ncol];
        v8f C = {0.f, 0.f, 0.f, 0.f, 0.f, 0.f, 0.f, 0.f};
        // 8 args: (neg_a, A, neg_b, B, c_mod, C, reuse_a, reuse_b)
        C = __builtin_amdgcn_wmma_f32_16x16x4_f32(
                false, A, false, B, (short)0, C, false, false);
        const float bias = b_s2f[ncol];
#pragma unroll
        for (int r = 0; r < 8; ++r) {
            // C VGPR r: lanes 0-15 -> M=r, lanes 16-31 -> M=r+8
            const int rowm = __shfl(myrow, r + 8 * hi, 32);
            __builtin_nontemporal_store(C[r] + bias,
                                        out + (size_t)rowm * D_MODEL + ncol);
        }
    }
}

extern "C" void kernel_launch(void* const* d_in, const int* in_sizes, int n_in,
                              void* d_out, int out_size, void* d_ws, size_t ws_size,
                              hipStream_t stream) {
    const float* x        = (const float*)d_in[0];
    const float* tgt      = (const float*)d_in[1];
    const int*   idx_a    = (const int*)d_in[2];
    const int*   idx_b    = (const int*)d_in[3];
    const float* W_proj_a = (const float*)d_in[4];
    const float* b_proj_a = (const float*)d_in[5];
    const float* W_proj_b = (const float*)d_in[6];
    const float* b_proj_b = (const float*)d_in[7];
    const float* W_f2s_a  = (const float*)d_in[8];
    const float* b_f2s_a  = (const float*)d_in[9];
    const float* W_f2s_b  = (const float*)d_in[10];
    const float* b_f2s_b  = (const float*)d_in[11];
    const float* W_s2f_a  = (const float*)d_in[12];
    const float* b_s2f_a  = (const float*)d_in[13];
    const float* W_s2f_b  = (const float*)d_in[14];
    const float* b_s2f_b  = (const float*)d_in[15];
    float* out = (float*)d_out;

    const int na = in_sizes[2];   // 131072
    const int nb = in_sizes[3];   // 131072

    dim3 block(256);              // 8 wave32s -> 128 agents per block
    agent_fused_kernel<0><<<dim3(na / 128), block, 0, stream>>>(
        x, tgt, idx_a, W_proj_a, b_proj_a, W_f2s_a, b_f2s_a, W_s2f_a, b_s2f_a, out);
    agent_fused_kernel<1><<<dim3(nb / 128), block, 0, stream>>>(
        x, tgt, idx_b, W_proj_b, b_proj_b, W_f2s_b, b_f2s_b, W_s2f_b, b_s2f_b, out);
}